// ScaledDotProductAttention_64544768524298
// MI455X (gfx1250) — compile-verified
//
#include <hip/hip_runtime.h>

typedef float v2f __attribute__((ext_vector_type(2)));
typedef float v8f __attribute__((ext_vector_type(8)));

#define B_   2
#define H_   16
#define S_   2048
#define DK_  64
#define BH_  (B_ * H_)
#define QTILES_ (S_ / 16)   // 128 query tiles of 16 rows per (b,h)

// One wave (32 threads) owns a 16-row query tile for one (b,h).
// Pass A: S = Q*K^T / 8 via v_wmma_f32_16x16x4_f32, online row (max,sum),
//         raw scaled scores written to attn buffer (scratch = final home).
// Pass B: re-read scores in A-fragment layout, p = exp(s-m)/l, write attn,
//         O += p * V via v_wmma_f32_16x16x4_f32.
__global__ __launch_bounds__(32) void attn_fwd_f32wmma(
    const float* __restrict__ Q, const float* __restrict__ K,
    const float* __restrict__ V, float* __restrict__ out,
    float* __restrict__ attn)
{
    const unsigned lane = threadIdx.x & 31u;
    const unsigned lo   = lane & 15u;   // row/col index within 16
    const unsigned hi   = lane >> 4;    // which half of the wave

    const unsigned tile = blockIdx.x;
    const unsigned bh   = tile >> 7;            // tile / QTILES_
    const unsigned m0   = (tile & 127u) * 16u;  // first query row of this tile

    const size_t qkv_base = (size_t)bh * S_ * DK_;

    // ---- Q tile as 16 A-fragments (K=4 each): lane -> row lo, dk = 4kc + 2*hi + v
    const float* qrow = Q + qkv_base + (size_t)(m0 + lo) * DK_ + 2u * hi;
    v2f qa[16];
#pragma unroll
    for (int kc = 0; kc < 16; ++kc)
        qa[kc] = *(const v2f*)(qrow + 4 * kc);

    float m[8], l[8];
#pragma unroll
    for (int r = 0; r < 8; ++r) { m[r] = -1.0e30f; l[r] = 0.0f; }

    float* attnW = attn + (size_t)bh * S_ * S_;

    // ================= Pass A: scores + online softmax stats =================
    for (int n0 = 0; n0 < S_; n0 += 16) {
        const float* krow = K + qkv_base + (size_t)(n0 + lo) * DK_ + 2u * hi;
        v8f c = {0.f, 0.f, 0.f, 0.f, 0.f, 0.f, 0.f, 0.f};
#pragma unroll
        for (int kc = 0; kc < 16; ++kc) {
            v2f kb = *(const v2f*)(krow + 4 * kc);   // B frag: row dk, col n=lo
            c = __builtin_amdgcn_wmma_f32_16x16x4_f32(
                    false, qa[kc], false, kb, (short)0, c, false, false);
        }
#pragma unroll
        for (int r = 0; r < 8; ++r) {
            float x = c[r] * 0.125f;                 // 1/sqrt(64)
            // row max across the 16 lanes holding this row
            float tm = x;
            tm = fmaxf(tm, __shfl_xor(tm, 1, 32));
            tm = fmaxf(tm, __shfl_xor(tm, 2, 32));
            tm = fmaxf(tm, __shfl_xor(tm, 4, 32));
            tm = fmaxf(tm, __shfl_xor(tm, 8, 32));
            float mnew = fmaxf(m[r], tm);
            float ts = __expf(x - mnew);
            ts += __shfl_xor(ts, 1, 32);
            ts += __shfl_xor(ts, 2, 32);
            ts += __shfl_xor(ts, 4, 32);
            ts += __shfl_xor(ts, 8, 32);
            l[r] = l[r] * __expf(m[r] - mnew) + ts;
            m[r] = mnew;
            // raw scaled score in C-layout position (64B-coalesced per row half)
            attnW[(size_t)(m0 + r + 8u * hi) * S_ + n0 + lo] = x;
        }
    }

    // ---- broadcast per-row m and 1/l to all lanes via LDS (per-wave, in-order)
    __shared__ float sm[16];
    __shared__ float si[16];
    if (lo == 0) {
#pragma unroll
        for (int r = 0; r < 8; ++r) {
            sm[r + 8u * hi] = m[r];
            si[r + 8u * hi] = 1.0f / l[r];
        }
    }
    __syncthreads();
    const float mA = sm[lo];   // stats for row q = lo (A-fragment row)
    const float iA = si[lo];

    // ================= Pass B: normalize attn + O = P * V =================
    v8f o[4];
#pragma unroll
    for (int j = 0; j < 4; ++j)
        o[j] = (v8f){0.f, 0.f, 0.f, 0.f, 0.f, 0.f, 0.f, 0.f};

    // A-layout view of this wave's score rows: lane -> row lo, col = n0+4kc2+2hi+v
    float* prow = attnW + (size_t)(m0 + lo) * S_ + 2u * hi;

    for (int n0 = 0; n0 < S_; n0 += 16) {
#pragma unroll
        for (int kc2 = 0; kc2 < 4; ++kc2) {
            v2f s = *(const v2f*)(prow + n0 + 4 * kc2);   // L2 hit (just written)
            v2f p;
            p.x = __expf(s.x - mA) * iA;
            p.y = __expf(s.y - mA) * iA;
            *(v2f*)(prow + n0 + 4 * kc2) = p;             // final attn values

            // V B-fragments: row krow = n0+4kc2+2hi+v, col d = 16j+lo
            const float* v0 = V + qkv_base + (size_t)(n0 + 4 * kc2 + 2u * hi) * DK_ + lo;
#pragma unroll
            for (int j = 0; j < 4; ++j) {
                v2f vb;
                vb.x = v0[16 * j];
                vb.y = v0[16 * j + DK_];
                o[j] = __builtin_amdgcn_wmma_f32_16x16x4_f32(
                           false, p, false, vb, (short)0, o[j], false, false);
            }
        }
    }

    // ---- store O tile: row r+8hi, col 16j+lo
    float* orow = out + qkv_base + (size_t)m0 * DK_;
#pragma unroll
    for (int j = 0; j < 4; ++j) {
#pragma unroll
        for (int r = 0; r < 8; ++r) {
            orow[(size_t)(r + 8u * hi) * DK_ + 16 * j + lo] = o[j][r];
        }
    }
}

extern "C" void kernel_launch(void* const* d_in, const int* in_sizes, int n_in,
                              void* d_out, int out_size, void* d_ws, size_t ws_size,
                              hipStream_t stream) {
    const float* Q = (const float*)d_in[0];
    const float* K = (const float*)d_in[1];
    const float* V = (const float*)d_in[2];
    float* out  = (float*)d_out;
    float* attn = out + (size_t)B_ * H_ * S_ * DK_;  // tuple: (out, attn) concatenated

    dim3 grid(BH_ * QTILES_);   // 4096 waves, each fully independent
    dim3 block(32);             // one wave32 per workgroup
    hipLaunchKernelGGL(attn_fwd_f32wmma, grid, block, 0, stream,
                       Q, K, V, out, attn);
}